// GazeKLDUnit_27891517620670
// MI455X (gfx1250) — compile-verified
//
#include <hip/hip_runtime.h>
#include <hip/hip_bf16.h>
#include <stdint.h>

// ---------------------------------------------------------------------------
// GazeKLD reduction, MI455X (gfx1250).
// Bandwidth-bound: 435 MB in -> 4 B out (~18.7 us floor at 23.3 TB/s).
// Pass 1 streams `true` through LDS with CDNA5 async global->LDS copies
// (global_load_async_to_lds_b128 / ASYNCcnt), DOUBLE-BUFFERED: the next
// tile's copies are in flight while the current tile is reduced out of LDS.
// Partial-counter waits (s_wait_asynccnt 0x19) exploit in-order async
// completion. Deterministic two-pass tree reduction (no float atomics).
// ---------------------------------------------------------------------------

#define TILE_ROWS      64
#define ROW_FLOATS     100                      // 2*T with T=50
#define ROW_F4         (ROW_FLOATS / 4)         // 25 float4 per row
#define TILE_F4        (TILE_ROWS * ROW_F4)     // 1600 float4 per tile
#define TILE_BYTES     (TILE_ROWS * ROW_FLOATS * 4)  // 25,600 B per buffer
#define BLOCK_THREADS  64                       // 2 waves (wave32)
#define F4_PER_THREAD  (TILE_F4 / BLOCK_THREADS)     // 25 async b128 / thread
#define PASS1_BLOCKS   4096
#define PASS2_THREADS  256

typedef __attribute__((address_space(3))) uint8_t lds_u8_t;

// Raw byte offset of an LDS object inside the workgroup's LDS allocation:
// generic -> addrspace(3) cast strips the aperture; ptrtoint of the 32-bit
// LDS pointer is exactly what GLOBAL_LOAD_ASYNC_TO_LDS expects in VDST.
__device__ __forceinline__ uint32_t lds_byte_offset(const void* p) {
    return (uint32_t)(uintptr_t)(const lds_u8_t*)p;
}

// Issue one tile's worth of async copies: 25 x b128 per thread, coalesced
// 512 B per wave-instruction. Global offsets are CLAMPED (not predicated) so
// every wave always issues exactly 25 instructions -> ASYNCcnt arithmetic
// for the partial waits below stays exact.
__device__ __forceinline__ void issue_tile_copy(uint64_t gbase,
                                                uint32_t ldsBufBase,
                                                uint32_t tileByte,
                                                uint32_t maxOff,
                                                int tid)
{
#pragma unroll
    for (int j = 0; j < F4_PER_THREAD; ++j) {
        const uint32_t li      = (uint32_t)(j * BLOCK_THREADS + tid) * 16u;
        const uint32_t ldsAddr = ldsBufBase + li;
        uint32_t       goff    = tileByte + li;
        goff = (goff < maxOff) ? goff : maxOff;   // stay in-bounds, keep issue count fixed
        asm volatile("global_load_async_to_lds_b128 %0, %1, %2"
                     :
                     : "v"(ldsAddr), "v"(goff), "s"(gbase)
                     : "memory");
    }
}

__global__ __launch_bounds__(BLOCK_THREADS)
void gaze_kld_pass1_async(const float* __restrict__ pred,
                          const float* __restrict__ truep,
                          float* __restrict__ partials,
                          int B, int numTiles, uint32_t totalTrueBytes,
                          float invT)
{
    __shared__ __align__(16) float tile[2][TILE_ROWS * ROW_FLOATS]; // 2x25,600 B
    __shared__ float red[BLOCK_THREADS];

    const int      tid     = threadIdx.x;
    const uint32_t ldsBase = lds_byte_offset(&tile[0][0]);
    const uint64_t gbase   = (uint64_t)(uintptr_t)truep;
    const uint32_t maxOff  = totalTrueBytes - 16u;
    const int      G       = (int)gridDim.x;
    const float2* __restrict__ pred2 = (const float2*)pred;

    float acc = 0.0f;
    int   buf = 0;
    int   t   = (int)blockIdx.x;

    // Prologue: start filling buffer 0.
    if (t < numTiles)
        issue_tile_copy(gbase, ldsBase, (uint32_t)t * (uint32_t)TILE_BYTES,
                        maxOff, tid);

    for (; t < numTiles; t += G) {
        const int tn = t + G;                    // uniform across the block
        if (tn < numTiles) {
            // Prefetch next tile into the other buffer (its previous readers
            // all passed the barrier at the end of the prior iteration).
            issue_tile_copy(gbase,
                            ldsBase + (uint32_t)(buf ^ 1) * (uint32_t)TILE_BYTES,
                            (uint32_t)tn * (uint32_t)TILE_BYTES, maxOff, tid);
            // 50 outstanding; async loads complete in order per wave, so
            // ASYNCcnt <= 25 means the CURRENT tile is fully in LDS while
            // the prefetch remains in flight.
            asm volatile("s_wait_asynccnt 0x19" ::: "memory");
        } else {
            asm volatile("s_wait_asynccnt 0x0" ::: "memory");
        }
        __syncthreads();

        // Reduce one row per thread out of LDS (25 x float4).
        const int row = t * TILE_ROWS + tid;
        if (row < B) {
            const float4* __restrict__ r4 =
                (const float4*)(&tile[buf][tid * ROW_FLOATS]);
            float s0 = 0.0f, s1 = 0.0f;
#pragma unroll
            for (int k = 0; k < ROW_F4; ++k) {
                float4 v = r4[k];
                s0 += v.x + v.z;   // channel 0 (even floats)
                s1 += v.y + v.w;   // channel 1 (odd floats)
            }
            const float2 p  = pred2[(size_t)row * 2];   // pred[row, 0:2]
            const float  d0 = s0 * invT - p.x;
            const float  d1 = s1 * invT - p.y;
            acc += 0.5f * (d0 * d0 + d1 * d1);
        }
        __syncthreads();   // all readers done before this buffer is refilled
        buf ^= 1;
    }

    // Deterministic block tree reduction.
    red[tid] = acc;
    __syncthreads();
#pragma unroll
    for (int s = BLOCK_THREADS / 2; s > 0; s >>= 1) {
        if (tid < s) red[tid] += red[tid + s];
        __syncthreads();
    }
    if (tid == 0) partials[blockIdx.x] = red[0];
}

// Fallback for T != 50 (not taken with the reference shapes).
__global__ __launch_bounds__(256)
void gaze_kld_pass1_generic(const float* __restrict__ pred,
                            const float* __restrict__ truep,
                            float* __restrict__ partials,
                            int B, int T)
{
    __shared__ float red[256];
    const float invT = 1.0f / (float)T;
    float acc = 0.0f;
    for (int row = blockIdx.x * blockDim.x + threadIdx.x; row < B;
         row += gridDim.x * blockDim.x) {
        const float2* __restrict__ r = (const float2*)(truep + (size_t)row * 2 * T);
        float s0 = 0.0f, s1 = 0.0f;
        for (int t = 0; t < T; ++t) { float2 v = r[t]; s0 += v.x; s1 += v.y; }
        const float d0 = s0 * invT - pred[(size_t)row * 4 + 0];
        const float d1 = s1 * invT - pred[(size_t)row * 4 + 1];
        acc += 0.5f * (d0 * d0 + d1 * d1);
    }
    red[threadIdx.x] = acc;
    __syncthreads();
    for (int s = 128; s > 0; s >>= 1) {
        if ((int)threadIdx.x < s) red[threadIdx.x] += red[threadIdx.x + s];
        __syncthreads();
    }
    if (threadIdx.x == 0) partials[blockIdx.x] = red[0];
}

__global__ __launch_bounds__(PASS2_THREADS)
void gaze_kld_pass2(const float* __restrict__ partials, int n, float invB,
                    float* __restrict__ out)
{
    __shared__ float red[PASS2_THREADS];
    float acc = 0.0f;
    for (int i = threadIdx.x; i < n; i += PASS2_THREADS) acc += partials[i];
    red[threadIdx.x] = acc;
    __syncthreads();
#pragma unroll
    for (int s = PASS2_THREADS / 2; s > 0; s >>= 1) {
        if ((int)threadIdx.x < s) red[threadIdx.x] += red[threadIdx.x + s];
        __syncthreads();
    }
    if (threadIdx.x == 0) out[0] = red[0] * invB;
}

extern "C" void kernel_launch(void* const* d_in, const int* in_sizes, int n_in,
                              void* d_out, int out_size, void* d_ws, size_t ws_size,
                              hipStream_t stream)
{
    const float* pred  = (const float*)d_in[0];   // (B, 4) f32
    const float* truep = (const float*)d_in[1];   // (B, T, 2) f32
    float*       out   = (float*)d_out;           // scalar f32
    float*       parts = (float*)d_ws;

    const int       B         = in_sizes[0] / 4;
    const long long trueElems = (long long)in_sizes[1];
    const int       T         = (int)(trueElems / (2LL * (long long)B));

    int blocks = PASS1_BLOCKS;

    if (T == 50) {
        const int numTiles = (B + TILE_ROWS - 1) / TILE_ROWS;
        if (blocks > numTiles) blocks = numTiles;
        const uint32_t totalTrueBytes =
            (uint32_t)((uint64_t)B * (uint64_t)ROW_FLOATS * 4ull);
        gaze_kld_pass1_async<<<blocks, BLOCK_THREADS, 0, stream>>>(
            pred, truep, parts, B, numTiles, totalTrueBytes, 1.0f / 50.0f);
    } else {
        gaze_kld_pass1_generic<<<blocks, 256, 0, stream>>>(
            pred, truep, parts, B, T);
    }

    gaze_kld_pass2<<<1, PASS2_THREADS, 0, stream>>>(
        parts, blocks, 1.0f / (float)B, out);
}